// HypAgg_53102975647846
// MI455X (gfx1250) — compile-verified
//
#include <hip/hip_runtime.h>
#include <hip/hip_bf16.h>
#include <math.h>

typedef float v2f __attribute__((ext_vector_type(2)));
typedef float v8f __attribute__((ext_vector_type(8)));

#define DD 256
#define EPSF 1e-7f
#define MINN 1e-15f
#define MAXN 1e6f

__device__ __forceinline__ float wred(float v) {
#pragma unroll
  for (int off = 16; off > 0; off >>= 1) v += __shfl_xor(v, off, 32);
  return v;
}

// ---------------- Kernel 1: per-node logmap0 -> xt, zero agg -----------------
__global__ void node_pre_kernel(const float* __restrict__ x,
                                float* __restrict__ xt,
                                float* __restrict__ agg, int N) {
  int wid  = (blockIdx.x * blockDim.x + threadIdx.x) >> 5;
  int lane = threadIdx.x & 31;
  if (wid >= N) return;
  const float* xp = x + (size_t)wid * DD;
  float xv[8];
  float s = 0.f;
#pragma unroll
  for (int t = 0; t < 8; ++t) { xv[t] = xp[lane + 32 * t]; s += xv[t] * xv[t]; }
  s = wred(s);
  float x0 = __shfl(xv[0], 0, 32);
  float yn = fmaxf(sqrtf(fmaxf(s - x0 * x0, 0.f)), MINN);   // ||y||, clipped
  float th = fmaxf(x0, 1.0f + EPSF);
  float r  = acoshf(th) / yn;                               // SQRTK == 1
  float* o = xt + (size_t)wid * DD;
  float* a = agg + (size_t)wid * DD;
#pragma unroll
  for (int t = 0; t < 8; ++t) {
    int d = lane + 32 * t;
    o[d] = (d == 0) ? 0.f : xv[t] * r;
    a[d] = 0.f;
  }
}

// ------------- Kernel 2: dual WMMA GEMM  P = xt@W1a + b1,  Q = xt@W1b --------
__global__ void gemm_pq_kernel(const float* __restrict__ A,      // N x 256 (xt)
                               const float* __restrict__ Wfull,  // 513 x 256 (att_w1)
                               const float* __restrict__ b1,
                               float* __restrict__ P,
                               float* __restrict__ Q, int N) {
  int wid  = (blockIdx.x * blockDim.x + threadIdx.x) >> 5;
  int lane = threadIdx.x & 31;
  int tiles = (N >> 4) << 4;  // (N/16) row tiles * 16 col tiles
  if (wid >= tiles) return;
  int m0 = (wid >> 4) << 4;
  int n0 = (wid & 15) << 4;
  int mr = m0 + (lane & 15);
  int nc = n0 + (lane & 15);
  int kk = (lane >> 4) << 1;  // 0 or 2
  const float* WP = Wfull;              // rows 0..255
  const float* WQ = Wfull + 256 * DD;   // rows 256..511
  v8f accP, accQ;
  float bp = b1[nc];
#pragma unroll
  for (int i = 0; i < 8; ++i) { accP[i] = bp; accQ[i] = 0.f; }
  const float* arow = A + (size_t)mr * DD + kk;
#pragma unroll 4
  for (int k0 = 0; k0 < DD; k0 += 4) {
    v2f a, bP, bQ;
    a.x = arow[k0];
    a.y = arow[k0 + 1];
    bP.x = WP[(k0 + kk) * DD + nc];
    bP.y = WP[(k0 + kk + 1) * DD + nc];
    bQ.x = WQ[(k0 + kk) * DD + nc];
    bQ.y = WQ[(k0 + kk + 1) * DD + nc];
    accP = __builtin_amdgcn_wmma_f32_16x16x4_f32(false, a, false, bP, (short)0, accP, false, false);
    accQ = __builtin_amdgcn_wmma_f32_16x16x4_f32(false, a, false, bQ, (short)0, accQ, false, false);
  }
  int mbase = m0 + ((lane >> 4) << 3);
#pragma unroll
  for (int i = 0; i < 8; ++i) {
    size_t idx = (size_t)(mbase + i) * DD + nc;
    P[idx] = accP[i];
    Q[idx] = accQ[i];
  }
}

// ------------- Kernels 4/5: generic WMMA GEMM  C = act(ascale*A @ W + b) -----
__global__ void gemm_one_kernel(const float* __restrict__ A,
                                const float* __restrict__ W,
                                const float* __restrict__ bias,
                                float* __restrict__ C,
                                int N, float ascale, int act_silu) {
  int wid  = (blockIdx.x * blockDim.x + threadIdx.x) >> 5;
  int lane = threadIdx.x & 31;
  int tiles = (N >> 4) << 4;
  if (wid >= tiles) return;
  int m0 = (wid >> 4) << 4;
  int n0 = (wid & 15) << 4;
  int mr = m0 + (lane & 15);
  int nc = n0 + (lane & 15);
  int kk = (lane >> 4) << 1;
  v8f acc;
  float bb = bias[nc];
#pragma unroll
  for (int i = 0; i < 8; ++i) acc[i] = bb;
  const float* arow = A + (size_t)mr * DD + kk;
#pragma unroll 4
  for (int k0 = 0; k0 < DD; k0 += 4) {
    v2f a, b;
    a.x = arow[k0] * ascale;
    a.y = arow[k0 + 1] * ascale;
    b.x = W[(k0 + kk) * DD + nc];
    b.y = W[(k0 + kk + 1) * DD + nc];
    acc = __builtin_amdgcn_wmma_f32_16x16x4_f32(false, a, false, b, (short)0, acc, false, false);
  }
  int mbase = m0 + ((lane >> 4) << 3);
#pragma unroll
  for (int i = 0; i < 8; ++i) {
    float v = acc[i];
    if (act_silu) v = v / (1.f + expf(-v));  // silu
    C[(size_t)(mbase + i) * DD + nc] = v;
  }
}

// ---------------- Kernel 3: wave-per-edge attention + logmap + scatter -------
__global__ void edge_kernel(const float* __restrict__ x,
                            const float* __restrict__ P,
                            const float* __restrict__ Q,
                            const float* __restrict__ distances,
                            const int* __restrict__ edges,
                            const float* __restrict__ edge_mask,
                            const float* __restrict__ w1c,  // att_w1 row 512
                            const float* __restrict__ w2,   // att_w2 (256)
                            const float* __restrict__ b2,   // att_b2 (1)
                            float* __restrict__ agg,
                            int E) {
  int wid  = (blockIdx.x * blockDim.x + threadIdx.x) >> 5;
  int lane = threadIdx.x & 31;
  if (wid >= E) return;
  int r = edges[wid];
  int c = edges[E + wid];
  float de = distances[wid];
  float em = edge_mask[wid];
  const float* xr = x + (size_t)r * DD;
  const float* xc = x + (size_t)c * DD;
  const float* pr = P + (size_t)r * DD;
  const float* qc = Q + (size_t)c * DD;
  float ar[8], ac_[8];
  float s_xx = 0.f, s_hw = 0.f;
#pragma unroll
  for (int t = 0; t < 8; ++t) {
    int d = lane + 32 * t;
    ar[t]  = xr[d];
    ac_[t] = xc[d];
    s_xx += ar[t] * ac_[t];
    float h = pr[d] + qc[d] + de * w1c[d];   // b1 already folded into P
    h = h / (1.f + expf(-h));                // silu
    s_hw += h * w2[d];
  }
  s_xx = wred(s_xx);
  s_hw = wred(s_hw);
  float x0r = __shfl(ar[0], 0, 32);
  float x0c = __shfl(ac_[0], 0, 32);
  float mdot = s_xx - 2.f * x0r * x0c;                       // mink_dot(xr,xc)
  float att  = em / (1.f + expf(-(s_hw + b2[0])));           // sigmoid * mask
  // logmap(xr, xc)
  float xy = fminf(mdot + 1.f, -EPSF) - 1.f;
  float u[8];
  float s_uu = 0.f, s_xu = 0.f;
#pragma unroll
  for (int t = 0; t < 8; ++t) {
    u[t] = ac_[t] + xy * ar[t];
    s_uu += u[t] * u[t];
    s_xu += ar[t] * u[t];
  }
  s_uu = wred(s_uu);
  s_xu = wred(s_xu);
  float u0    = x0c + xy * x0r;
  float normu = sqrtf(fmaxf(s_uu - 2.f * u0 * u0, EPSF));
  float th    = fmaxf(-mdot, 1.f + EPSF);
  float acsh  = acoshf(th);
  float dist  = sqrtf(fminf(acsh * acsh, 50.f));
  float scl   = dist / normu;
  float ux    = (s_xu - x0r * u0) * scl;   // sum over d>=1 of xr*t_vec
  float res0  = ux / fmaxf(x0r, EPSF);
  float* ag = agg + (size_t)r * DD;
#pragma unroll
  for (int t = 0; t < 8; ++t) {
    int d = lane + 32 * t;
    float v = scl * u[t];
    if (d == 0) v = res0;
    atomicAdd(&ag[d], v * att);
  }
}

// --------- Kernel 6: self-tangent + mlp, proj_tan, expmap -> output ----------
__global__ void node_final_kernel(const float* __restrict__ x,
                                  const float* __restrict__ m2,
                                  float* __restrict__ out, int N) {
  int wid  = (blockIdx.x * blockDim.x + threadIdx.x) >> 5;
  int lane = threadIdx.x & 31;
  if (wid >= N) return;
  const float* xp = x + (size_t)wid * DD;
  const float* mp = m2 + (size_t)wid * DD;
  float xv[8], mv[8];
  float s = 0.f;
#pragma unroll
  for (int t = 0; t < 8; ++t) {
    int d = lane + 32 * t;
    xv[t] = xp[d];
    mv[t] = mp[d];
    s += xv[t] * xv[t];
  }
  s = wred(s);
  float x0 = __shfl(xv[0], 0, 32);
  // logmap(x, x):  u = (1+xy) * x
  float mdot = s - 2.f * x0 * x0;
  float xy   = fminf(mdot + 1.f, -EPSF) - 1.f;
  float f    = 1.f + xy;
  float u0   = f * x0;
  float s_uu = f * f * s;
  float normu = sqrtf(fmaxf(s_uu - 2.f * u0 * u0, EPSF));
  float th    = fmaxf(-mdot, 1.f + EPSF);
  float acsh  = acoshf(th);
  float dist  = sqrtf(fminf(acsh * acsh, 50.f));
  float scl   = dist / normu;
  float ux    = (f * s - x0 * u0) * scl;
  float self0 = ux / fmaxf(x0, EPSF);
  // out = self_tangent + mlp_out
  float ov[8];
  float s_xo = 0.f;
#pragma unroll
  for (int t = 0; t < 8; ++t) {
    int d = lane + 32 * t;
    float sv = (d == 0) ? self0 : scl * f * xv[t];
    ov[t] = sv + mv[t];
    s_xo += xv[t] * ov[t];
  }
  s_xo = wred(s_xo);
  float o0  = __shfl(ov[0], 0, 32);
  float ux2 = s_xo - x0 * o0;          // sum over d>=1 of x*out
  float sp0 = ux2 / fmaxf(x0, EPSF);   // proj_tan -> support_t[0]
  // expmap(support_t, x)
  float s_ss = 0.f;
#pragma unroll
  for (int t = 0; t < 8; ++t) {
    int d = lane + 32 * t;
    float sv = (d == 0) ? sp0 : ov[t];
    s_ss += sv * sv;
  }
  s_ss = wred(s_ss);
  float mds  = s_ss - 2.f * sp0 * sp0;
  float nrm  = fminf(sqrtf(fmaxf(mds, EPSF)), MAXN);
  float th2  = fmaxf(nrm, MINN);
  float ch   = coshf(th2);
  float shf  = sinhf(th2) / th2;
  float pv[8];
  float s_py = 0.f;
#pragma unroll
  for (int t = 0; t < 8; ++t) {
    int d = lane + 32 * t;
    float sv = (d == 0) ? sp0 : ov[t];
    pv[t] = ch * xv[t] + shf * sv;
    if (d != 0) s_py += pv[t] * pv[t];
  }
  s_py = wred(s_py);
  float out0 = sqrtf(1.f + s_py);      // proj
  float* op = out + (size_t)wid * DD;
#pragma unroll
  for (int t = 0; t < 8; ++t) {
    int d = lane + 32 * t;
    op[d] = (d == 0) ? out0 : pv[t];
  }
}

extern "C" void kernel_launch(void* const* d_in, const int* in_sizes, int n_in,
                              void* d_out, int out_size, void* d_ws, size_t ws_size,
                              hipStream_t stream) {
  const float* x         = (const float*)d_in[0];
  const float* distances = (const float*)d_in[1];
  const int*   edges     = (const int*)d_in[2];
  // d_in[3] node_mask: unused by reference math
  const float* edge_mask = (const float*)d_in[4];
  const float* att_w1    = (const float*)d_in[5];
  const float* att_b1    = (const float*)d_in[6];
  const float* att_w2    = (const float*)d_in[7];
  const float* att_b2    = (const float*)d_in[8];
  const float* mlp_w1    = (const float*)d_in[9];
  const float* mlp_b1    = (const float*)d_in[10];
  const float* mlp_w2    = (const float*)d_in[11];
  const float* mlp_b2    = (const float*)d_in[12];

  int N = in_sizes[0] / DD;   // 10000
  int E = in_sizes[1];        // 320000

  size_t nd = (size_t)N * DD;
  float* ws  = (float*)d_ws;
  float* xt  = ws;
  float* P   = ws + 1 * nd;
  float* Q   = ws + 2 * nd;
  float* agg = ws + 3 * nd;
  float* H   = ws + 4 * nd;
  float* M2  = ws + 5 * nd;

  dim3 blk(256);
  int blocksN = (N * 32 + 255) / 256;                 // one wave per node
  node_pre_kernel<<<blocksN, blk, 0, stream>>>(x, xt, agg, N);

  int tiles   = (N / 16) * 16;                        // 16x16 output tiles
  int blocksG = (tiles * 32 + 255) / 256;
  gemm_pq_kernel<<<blocksG, blk, 0, stream>>>(xt, att_w1, att_b1, P, Q, N);

  long long ethreads = (long long)E * 32;             // one wave per edge
  int blocksE = (int)((ethreads + 255) / 256);
  edge_kernel<<<blocksE, blk, 0, stream>>>(x, P, Q, distances, edges, edge_mask,
                                           att_w1 + 512 * DD, att_w2, att_b2,
                                           agg, E);

  // H = silu(agg/100 @ mlp_w1 + b1)
  gemm_one_kernel<<<blocksG, blk, 0, stream>>>(agg, mlp_w1, mlp_b1, H, N, 0.01f, 1);
  // M2 = H @ mlp_w2 + b2
  gemm_one_kernel<<<blocksG, blk, 0, stream>>>(H, mlp_w2, mlp_b2, M2, N, 1.0f, 0);

  node_final_kernel<<<blocksN, blk, 0, stream>>>(x, M2, (float*)d_out, N);
}